// TopKRouter_84318797955293
// MI455X (gfx1250) — compile-verified
//
#include <hip/hip_runtime.h>
#include <math.h>

typedef __attribute__((ext_vector_type(2))) float v2f;
typedef __attribute__((ext_vector_type(8))) float v8f;

#define HID   4096
#define NEXP  64
#define NTOK  16384   // B*S = 4*4096
#define KSPL  (HID / 4)   // K slice per wave = 1024

// ---------------------------------------------------------------------------
// Kernel 1: logits = h @ W^T + b via V_WMMA_F32_16X16X4_F32, split-K in-block.
// Block = 128 threads = 4 waves, all on the SAME 16 tokens x 64 experts.
// Wave w accumulates K range [w*1024, (w+1)*1024); partials reduced via LDS.
// Grid = 16384/16 = 1024 blocks -> 4096 waves (4x the previous concurrency),
// while h is still streamed exactly once.
// ---------------------------------------------------------------------------
__global__ __launch_bounds__(128) void gemm_logits_kernel(
    const float* __restrict__ h, const float* __restrict__ W,
    const float* __restrict__ bias, float* __restrict__ logits) {
  __shared__ float s_acc[4][4][32][8];   // [wave][n-tile][lane][v] = 16 KB

  const int wave = threadIdx.x >> 5;
  const int lane = threadIdx.x & 31;
  const int l16  = lane & 15;
  const int grp  = lane >> 4;                 // 0: K=0,1  1: K=2,3
  const int tokBase = blockIdx.x * 16;
  const int kOff = wave * KSPL;

  // A fragment: h[(tokBase+l16)*H + kOff + k + 2*grp], float2 per lane per step
  const float* __restrict__ ha = h + (size_t)(tokBase + l16) * HID + kOff + (grp << 1);
  // B fragment: W[(n*16+l16)*H + kOff + k + 2*grp], float2 per lane per step
  const float* __restrict__ wa = W + (size_t)l16 * HID + kOff + (grp << 1);

  v8f acc[4] = {v8f{0.f}, v8f{0.f}, v8f{0.f}, v8f{0.f}};

#pragma unroll 8
  for (int k = 0; k < KSPL; k += 4) {
    v2f a = *(const v2f*)(ha + k);
#pragma unroll
    for (int n = 0; n < 4; ++n) {
      v2f bv = *(const v2f*)(wa + (size_t)(n * 16) * HID + k);
      // 8 args: (neg_a, A, neg_b, B, c_mod, C, reuse_a, reuse_b)
      acc[n] = __builtin_amdgcn_wmma_f32_16x16x4_f32(
          false, a, false, bv, (short)0, acc[n], false, false);
    }
  }

  // Spill partial C tiles to LDS (8 consecutive floats per lane per tile)
#pragma unroll
  for (int n = 0; n < 4; ++n)
#pragma unroll
    for (int v = 0; v < 8; ++v)
      s_acc[wave][n][lane][v] = acc[n][v];
  __syncthreads();

  // Cross-wave reduction: wave w finalizes n-tile w.
  // D layout: vgpr v -> (M = v + 8*grp, N = l16)
  const int n = wave;
  const float bb = bias[n * 16 + l16];
#pragma unroll
  for (int v = 0; v < 8; ++v) {
    float s = s_acc[0][n][lane][v] + s_acc[1][n][lane][v] +
              s_acc[2][n][lane][v] + s_acc[3][n][lane][v];
    const int tok = tokBase + v + (grp << 3);
    logits[(size_t)tok * NEXP + n * 16 + l16] = s + bb;
  }
}

// ---------------------------------------------------------------------------
// Kernel 2: per-token top-2 + softmax-of-2 -> route_weights; accumulate
// importance (full softmax sums) and load (top-2 hit counts) via LDS + atomics
// ---------------------------------------------------------------------------
__global__ __launch_bounds__(256) void route_kernel(
    const float* __restrict__ logits, float* __restrict__ rw,
    float* __restrict__ imp, float* __restrict__ loadc) {
  __shared__ float s_imp[NEXP];
  __shared__ float s_load[NEXP];
  if (threadIdx.x < NEXP) { s_imp[threadIdx.x] = 0.f; s_load[threadIdx.x] = 0.f; }
  __syncthreads();

  const int tok = blockIdx.x * 256 + threadIdx.x;
  const float* __restrict__ lr = logits + (size_t)tok * NEXP;
  float* __restrict__ rr = rw + (size_t)tok * NEXP;

  // top-2 scan; strict '>' keeps the lowest index on ties (matches lax.top_k)
  float v1 = -3.0e38f; int i1 = 0;
  float v2 = -3.0e38f; int i2 = 0;
  for (int e = 0; e < NEXP; ++e) {
    float x = lr[e];
    if (x > v1)      { v2 = v1; i2 = i1; v1 = x; i1 = e; }
    else if (x > v2) { v2 = x;  i2 = e; }
  }

  // stable softmax over the two selected logits
  const float e2   = expf(v2 - v1);
  const float inv2 = 1.0f / (1.0f + e2);
  const float s1 = inv2;
  const float s2 = e2 * inv2;

  // full softmax denominator (max is v1 by construction)
  float sum = 0.f;
  for (int e = 0; e < NEXP; ++e) sum += expf(lr[e] - v1);
  const float invs = 1.0f / sum;

  for (int e = 0; e < NEXP; ++e) {
    const float p = expf(lr[e] - v1) * invs;
    atomicAdd(&s_imp[e], p);
    rr[e] = (e == i1) ? s1 : ((e == i2) ? s2 : 0.0f);
  }
  atomicAdd(&s_load[i1], 1.0f);
  atomicAdd(&s_load[i2], 1.0f);
  __syncthreads();

  if (threadIdx.x < NEXP) {
    atomicAdd(&imp[threadIdx.x],   s_imp[threadIdx.x]);
    atomicAdd(&loadc[threadIdx.x], s_load[threadIdx.x]);
  }
}

// ---------------------------------------------------------------------------
// Kernel 3: aux_loss = mean_e( importance[e] * (load[e]/(B*S)) * E^2 )
// ---------------------------------------------------------------------------
__global__ __launch_bounds__(64) void finalize_kernel(
    const float* __restrict__ imp, const float* __restrict__ loadc,
    float* __restrict__ aux_out) {
  __shared__ float s[NEXP];
  const int e = threadIdx.x;
  s[e] = imp[e] * (loadc[e] / (float)NTOK) * (float)(NEXP * NEXP);
  __syncthreads();
  for (int o = 32; o > 0; o >>= 1) {
    if (e < o) s[e] += s[e + o];
    __syncthreads();
  }
  if (e == 0) aux_out[0] = s[0] / (float)NEXP;
}

__global__ void zero_kernel(float* __restrict__ p, int n) {
  int i = blockIdx.x * blockDim.x + threadIdx.x;
  if (i < n) p[i] = 0.f;
}

// ---------------------------------------------------------------------------
extern "C" void kernel_launch(void* const* d_in, const int* in_sizes, int n_in,
                              void* d_out, int out_size, void* d_ws, size_t ws_size,
                              hipStream_t stream) {
  (void)in_sizes; (void)n_in; (void)out_size; (void)ws_size;
  const float* h = (const float*)d_in[0];   // [4,4096,4096]
  const float* W = (const float*)d_in[1];   // [64,4096]
  const float* b = (const float*)d_in[2];   // [64]
  // d_in[3] is k==2 (hardcoded)

  float* rw  = (float*)d_out;                      // [16384, 64]
  float* aux = rw + (size_t)NTOK * NEXP;           // scalar

  float* logits = (float*)d_ws;                    // 16384*64 floats = 4 MiB
  float* imp    = logits + (size_t)NTOK * NEXP;    // [64]
  float* loadc  = imp + NEXP;                      // [64]

  zero_kernel<<<1, 128, 0, stream>>>(imp, 2 * NEXP);
  gemm_logits_kernel<<<NTOK / 16, 128, 0, stream>>>(h, W, b, logits);
  route_kernel<<<NTOK / 256, 256, 0, stream>>>(logits, rw, imp, loadc);
  finalize_kernel<<<1, 64, 0, stream>>>(imp, loadc, aux);
}